// CCLNas_25898652795266
// MI455X (gfx1250) — compile-verified
//
#include <hip/hip_runtime.h>
#include <hip/hip_bf16.h>
#include <math.h>

// ---- problem constants (match reference) ----
#define EDIM   512
#define DDIM   32
#define T_INV  (1.0f / 0.07f)
#define MARGIN 2.0f
#define MIN_NEG 4500
#define L_COLS (1 + MIN_NEG)

typedef __attribute__((ext_vector_type(2))) float v2f;
typedef __attribute__((ext_vector_type(8))) float v8f;

// Kernel 1: per-row inverse norm of q  (one wave32 per row, D==32==warpSize)
__global__ void ccl_qinvnorm_kernel(const float* __restrict__ q,
                                    float* __restrict__ invnorm, int N) {
    int gw   = (blockIdx.x * blockDim.x + threadIdx.x) >> 5;   // global wave id
    int lane = threadIdx.x & 31;
    if (gw >= N) return;                                       // whole-wave uniform
    float v = q[(size_t)gw * DDIM + lane];
    float s = v * v;
    for (int off = 16; off; off >>= 1) s += __shfl_xor(s, off, 32);
    if (lane == 0) invnorm[gw] = rsqrtf(s);
}

// Kernel 2: one workgroup (32 waves) per sample; d[] lives in LDS (160KB of the
// 320KB/WGP pool), negatives selected by stable ballot-scan, dots via WMMA f32.
__global__ __launch_bounds__(1024)
void ccl_sample_kernel(const float* __restrict__ P,
                       const float* __restrict__ q,
                       const int*   __restrict__ sid,
                       const float* __restrict__ invnorm,
                       float* __restrict__ out, int N, int S) {
    extern __shared__ char smem_raw[];
    float* d_lds   = (float*)smem_raw;                    // N floats (160000 B)
    float* p_i     = (float*)(smem_raw + (size_t)N * 4);  // EDIM floats
    int*   neg_sel = (int*)(p_i + EDIM);                  // MIN_NEG ints
    float* redV    = (float*)(neg_sel + MIN_NEG);         // 1024 floats
    int*   redI    = (int*)(redV + 1024);                 // 1024 ints
    float* cacc    = (float*)(redI + 1024);               // 32 floats
    float* cN      = cacc + 32;                           // 32 floats
    int*   waveTot = (int*)(cN + 32);                     // 32 ints
    int*   wavePre = waveTot + 32;                        // 32 ints
    float* s_min   = (float*)(wavePre + 32);
    int*   s_j     = (int*)(s_min + 1);
    int*   s_base  = s_j + 1;
    int*   s_chunk = s_base + 1;

    const int tid  = threadIdx.x;
    const int lane = tid & 31;
    const int wid  = tid >> 5;
    const int s    = blockIdx.x;
    const int i    = sid[s];

    // stage the sample's encoding row into LDS
    for (int e = tid; e < EDIM; e += blockDim.x)
        p_i[e] = P[(size_t)i * EDIM + e];
    if (tid == 0) *s_base = 0;
    __syncthreads();

    // ---- pass 1: L1 distances, one wave per row (coalesced 2KB/row) ----
    for (int n = wid; n < N; n += 32) {
        const float4* row = (const float4*)(P + (size_t)n * EDIM);
        float sum = 0.f;
#pragma unroll
        for (int jj = 0; jj < 4; ++jj) {
            float4 a = row[lane * 4 + jj];
            sum += fabsf(a.x - p_i[lane * 16 + jj * 4 + 0]);
            sum += fabsf(a.y - p_i[lane * 16 + jj * 4 + 1]);
            sum += fabsf(a.z - p_i[lane * 16 + jj * 4 + 2]);
            sum += fabsf(a.w - p_i[lane * 16 + jj * 4 + 3]);
        }
        for (int off = 16; off; off >>= 1) sum += __shfl_xor(sum, off, 32);
        if (lane == 0) d_lds[n] = sum;
    }
    __syncthreads();
    if (tid == 0) d_lds[i] = 1e30f;     // exclude self from min (ref: d[i]=100)
    __syncthreads();

    // ---- min + first-argmin over LDS ----
    float bv = 1e30f; int bi = N;
    for (int n = tid; n < N; n += 1024) {
        float v = d_lds[n];
        if (v < bv) { bv = v; bi = n; }  // ascending n -> first occurrence kept
    }
    redV[tid] = bv; redI[tid] = bi;
    __syncthreads();
    for (int st = 512; st >= 1; st >>= 1) {
        if (tid < st) {
            float ov = redV[tid + st]; int oi = redI[tid + st];
            if (ov < redV[tid] || (ov == redV[tid] && oi < redI[tid])) {
                redV[tid] = ov; redI[tid] = oi;
            }
        }
        __syncthreads();
    }
    if (tid == 0) { *s_min = redV[0]; *s_j = redI[0]; d_lds[i] = 0.0f; } // ref: d[i]=0
    if (tid < DDIM) cacc[tid] = 0.0f;
    __syncthreads();
    const float minv = *s_min;
    const int   jpos = *s_j;

    // ---- center = normalize(sum of q-hat rows where d == minv) ----
    for (int n = tid; n < N; n += 1024) {
        if (d_lds[n] == minv) {
            float inrm = invnorm[n];
            for (int e = 0; e < DDIM; ++e)
                atomicAdd(&cacc[e], q[(size_t)n * DDIM + e] * inrm);
        }
    }
    __syncthreads();
    if (tid < 32) {
        float c  = cacc[lane];
        float sq = c * c;
        for (int off = 16; off; off >>= 1) sq += __shfl_xor(sq, off, 32);
        float cn = c * rsqrtf(sq);        // cnt cancels under normalization
        cN[lane] = cn;
        // positive pair logit
        float pj = q[(size_t)jpos * DDIM + lane] * invnorm[jpos];
        float pp = pj * cn;
        for (int off = 16; off; off >>= 1) pp += __shfl_xor(pp, off, 32);
        if (lane == 0) out[(size_t)s * L_COLS] = pp * T_INV;
        // centers output block
        out[(size_t)S * L_COLS + S + (size_t)s * DDIM + lane] = cn;
    }
    if (tid == 0) out[(size_t)S * L_COLS + s] = 0.0f;   // label (int32 0 bits)
    __syncthreads();

    // ---- stable selection of first MIN_NEG negatives (ascending index) ----
    const float thr = minv + MARGIN;
    const int nChunks = (N + 1023) / 1024;
    for (int phase = 0; phase < 2; ++phase) {
        if (*s_base >= MIN_NEG) break;            // uniform (behind barriers)
        for (int c = 0; c < nChunks; ++c) {
            int n = c * 1024 + tid;
            bool m;
            if (phase == 0) m = (n < N) && (d_lds[n] >= thr);
            else            m = (n < N) && !(d_lds[n] >= thr); // ref argsort tail
            unsigned long long bal = __ballot(m);
            int lanePre = __popcll(bal & ((1ull << lane) - 1ull));
            int wTot    = __popcll(bal);
            if (lane == 0) waveTot[wid] = wTot;
            __syncthreads();
            if (tid < 32) {
                int ex = 0;
                for (int k = 0; k < lane; ++k) ex += waveTot[k];
                wavePre[lane] = ex;
                if (lane == 31) *s_chunk = ex + waveTot[31];
            }
            __syncthreads();
            int rank = *s_base + wavePre[wid] + lanePre;
            if (m && rank < MIN_NEG) neg_sel[rank] = n;
            __syncthreads();
            if (tid == 0) *s_base += *s_chunk;
            __syncthreads();
            if (*s_base >= MIN_NEG) break;        // uniform
        }
    }
    __syncthreads();

    // ---- negative logits via V_WMMA_F32_16X16X4_F32 (K=32 in 8 chained steps)
    // A: 16 gathered q rows x 4 dims; B: center broadcast to all 16 columns.
    // A f32 16x4 layout: lane l holds row (l&15); VGPR0/1 = K {0,1} (l<16) or {2,3}.
    const int nTiles = (MIN_NEG + 15) / 16;       // 282
    const int half = (lane < 16) ? 0 : 2;
    const int r    = lane & 15;
    for (int t = wid; t < nTiles; t += 32) {      // wave-uniform: EXEC all ones
        int rr  = t * 16 + r;
        int row = (rr < MIN_NEG) ? neg_sel[rr] : 0;  // tail pad, masked at write
        const float* qrow = q + (size_t)row * DDIM;
        v8f c = {};
#pragma unroll
        for (int kb = 0; kb < 8; ++kb) {
            v2f a, b;
            a.x = qrow[kb * 4 + half];
            a.y = qrow[kb * 4 + half + 1];
            b.x = cN[kb * 4 + half];
            b.y = cN[kb * 4 + half + 1];
            c = __builtin_amdgcn_wmma_f32_16x16x4_f32(
                    false, a, false, b, (short)0, c, false, false);
        }
        // D layout: column N=0 lives in lane 0 (M=0..7) and lane 16 (M=8..15)
        if (r == 0) {
            int mb = (lane < 16) ? 0 : 8;
#pragma unroll
            for (int v = 0; v < 8; ++v) {
                int orow = t * 16 + mb + v;
                if (orow < MIN_NEG) {
                    float cv = c[v];
                    out[(size_t)s * L_COLS + 1 + orow] =
                        cv * invnorm[neg_sel[orow]] * T_INV;
                }
            }
        }
    }
}

extern "C" void kernel_launch(void* const* d_in, const int* in_sizes, int n_in,
                              void* d_out, int out_size, void* d_ws, size_t ws_size,
                              hipStream_t stream) {
    const float* P   = (const float*)d_in[0];   // [N, 512] f32
    const float* q   = (const float*)d_in[1];   // [N, 32]  f32
    const int*   sid = (const int*)d_in[2];     // [S]      i32
    const int N = in_sizes[1] / DDIM;
    const int S = in_sizes[2];

    float* invnorm = (float*)d_ws;              // N floats (160 KB) scratch

    // kernel 1: per-row inverse norms
    const int thr1 = 256;                       // 8 waves/block
    const int blocks1 = (N + (thr1 / 32) - 1) / (thr1 / 32);
    ccl_qinvnorm_kernel<<<blocks1, thr1, 0, stream>>>(q, invnorm, N);

    // kernel 2: one workgroup per sample; dynamic LDS holds d[N] + scan state
    size_t smem = (size_t)N * 4          // d
                + EDIM * 4               // p_i
                + MIN_NEG * 4            // neg_sel
                + 1024 * 4 * 2           // redV/redI
                + 32 * 4 * 4             // cacc/cN/waveTot/wavePre
                + 64;                    // scalars + pad
    ccl_sample_kernel<<<S, 1024, smem, stream>>>(P, q, sid, invnorm,
                                                 (float*)d_out, N, S);
}